// SES_84722524881463
// MI455X (gfx1250) — compile-verified
//
#include <hip/hip_runtime.h>
#include <cmath>

typedef __attribute__((ext_vector_type(2))) float v2f;
typedef __attribute__((ext_vector_type(8))) float v8f;

#define BLOCK_THREADS 256
#define WAVES_PER_BLOCK (BLOCK_THREADS / 32)
#define TILE 256   // 16 segments x 16 elements per wave

// (1-a)^d for lane-dependent small d, via v_exp_f32(d * v_log_f32(1-a)).
// Exact for d==0; for oma==0: log2->-inf, exp2(d*-inf)=0 for d>0 (correct).
__device__ inline float pow_dyn(float l2oma, int d) {
    return (d == 0) ? 1.0f : __builtin_amdgcn_exp2f((float)d * l2oma);
}

// ---------------------------------------------------------------------------
// Within-tile scan via WMMA: P = M * B for 16 independent 16-elem segments.
//   M[i][k] = (1-a)^(i-k) for i>=k else 0   (16x16 lower triangular)
//   B[k][n] = a * x[base + n*16 + k]        (column n = segment n)
// Four chained V_WMMA_F32_16X16X4_F32. D[i][n] = inclusive scan value at
// position i of segment n with zero carry-in.
// A layout (16x4 f32): lane L -> row m=L&15; v0 holds K=2*(L>=16), v1 K+1.
// B layout (4x16 f32): lane L -> col n=L&15; v0 holds K=2*(L>=16), v1 K+1.
// D layout: vgpr r -> row r + 8*(L>=16), col L&15.
// ---------------------------------------------------------------------------
__device__ inline v8f tile_scan_wmma(const float* __restrict__ x, long long base,
                                     float a, float l2oma) {
    const int lane = __lane_id();
    const int m    = lane & 15;            // A row / B column for this lane
    const int koff = (lane >> 4) * 2;      // K offset within each 4-slice

    v8f c = {};
#pragma unroll
    for (int kb = 0; kb < 4; ++kb) {
        const int k0 = kb * 4 + koff;
        v2f av;
        av.x = (m >= k0)     ? pow_dyn(l2oma, m - k0)     : 0.0f;
        av.y = (m >= k0 + 1) ? pow_dyn(l2oma, m - k0 - 1) : 0.0f;
        const v2f xv = *reinterpret_cast<const v2f*>(x + base + (long long)m * 16 + k0);
        const v2f bv = xv * a;
        c = __builtin_amdgcn_wmma_f32_16x16x4_f32(
                /*neg_a=*/false, av, /*neg_b=*/false, bv,
                /*c_mod=*/(short)0, c, /*reuse_a=*/false, /*reuse_b=*/false);
    }
    return c;
}

// ---------------------------------------------------------------------------
// Phase 1: one 256-element tile per wave -> tile aggregate
//   B_tile = fold over segments: c = (1-a)^16 * c + P[15][s],  c0 = 0
// P[15][s] lives in d[7], lanes 16..31 (row 15), column s = lane-16.
// ---------------------------------------------------------------------------
__global__ void ses_phase1(const float* __restrict__ x,
                           const float* __restrict__ alpha,
                           float* __restrict__ btile, int ntiles) {
    const int wave = blockIdx.x * WAVES_PER_BLOCK + (threadIdx.x >> 5);
    if (wave >= ntiles) return;                    // wave-uniform guard

    const float a     = alpha[0];
    const float oma   = 1.0f - a;
    const float l2oma = __builtin_amdgcn_logf(oma);
    const long long base = (long long)wave * TILE;

    __builtin_prefetch(x + base, 0, 1);            // global_prefetch_b8
    v8f d = tile_scan_wmma(x, base, a, l2oma);

    // q = (1-a)^16 by squaring (robust for oma==0)
    float q = oma * oma;  q *= q;  q *= q;  q *= q;

    const float last = d[7];
    float c = 0.0f;
#pragma unroll
    for (int s = 0; s < 16; ++s) {
        float v = __shfl(last, 16 + s, 32);        // P[15][s]
        c = q * c + v;
    }
    if (__lane_id() == 0) btile[wave] = c;
}

// ---------------------------------------------------------------------------
// Phase 2: single wave scans the ntiles aggregates -> per-tile carry level.
// ---------------------------------------------------------------------------
__global__ void ses_phase2(const float* __restrict__ btile,
                           const float* __restrict__ level,
                           const float* __restrict__ alpha,
                           float* __restrict__ carry, int ntiles) {
    const int lane = __lane_id();
    const int tpl  = ntiles / 32;                  // tiles per lane
    const float a   = alpha[0];
    const float oma = 1.0f - a;

    // A256 = (1-a)^256 by squaring
    float A256 = oma;
#pragma unroll
    for (int i = 0; i < 8; ++i) A256 *= A256;

    const int t0 = lane * tpl;
    float B = 0.0f;
    for (int t = 0; t < tpl; ++t)                  // lane aggregate
        B = A256 * B + btile[t0 + t];

    // Alane = A256^tpl by exponentiation-by-squaring (uniform scalar loop)
    float Alane = 1.0f, pb = A256;
    for (int e = tpl; e; e >>= 1) { if (e & 1) Alane *= pb; pb *= pb; }

    float c = level[0];                            // exclusive scan over lanes
#pragma unroll
    for (int m = 0; m < 31; ++m) {
        float Bm = __shfl(B, m, 32);
        if (m < lane) c = Alane * c + Bm;
    }

    float cr = c;                                  // write per-tile carries
    for (int t = 0; t < tpl; ++t) {
        carry[t0 + t] = cr;
        cr = A256 * cr + btile[t0 + t];
    }
}

// ---------------------------------------------------------------------------
// Phase 3: redo the tile WMMA scan, apply carries, emit forecasts.
//   out[base + n*16 + i] = (1-a)^i * C_n + (i>0 ? P[i-1][n] : 0)
// where C_n = carry-in level of segment n inside this tile.
// ---------------------------------------------------------------------------
__global__ void ses_phase3(const float* __restrict__ x,
                           const float* __restrict__ alpha,
                           const float* __restrict__ carry,
                           float* __restrict__ out, int ntiles) {
    const int wave = blockIdx.x * WAVES_PER_BLOCK + (threadIdx.x >> 5);
    if (wave >= ntiles) return;                    // wave-uniform guard

    const int lane = __lane_id();
    const int m    = lane & 15;                    // this lane's column n
    const bool hi  = lane >= 16;                   // rows 8..15 vs 0..7

    const float a     = alpha[0];
    const float oma   = 1.0f - a;
    const float l2oma = __builtin_amdgcn_logf(oma);
    const long long base = (long long)wave * TILE;

    __builtin_prefetch(x + base, 0, 1);
    v8f d = tile_scan_wmma(x, base, a, l2oma);

    // q = (1-a)^16 and oma^8 by squaring
    float oma2 = oma * oma, oma4 = oma2 * oma2, oma8 = oma4 * oma4;
    float q = oma8 * oma8;

    // Segment carry C_n: fold carry_in through segments 0..n-1 with factor q.
    const float last = d[7];                       // rows 7 (lo) / 15 (hi)
    float c = carry[wave];
#pragma unroll
    for (int s = 0; s < 15; ++s) {
        float v = __shfl(last, 16 + s, 32);        // P[15][s]
        if (s < m) c = q * c + v;
    }

    // Row-7 value of this column (prev-row for row 8 on high lanes).
    const float p7 = __shfl(d[7], m, 32);          // low lanes hold row 7

    float res[8];
    float pw = hi ? oma8 : 1.0f;                   // (1-a)^i running power
#pragma unroll
    for (int r = 0; r < 8; ++r) {
        float prev;
        if (r == 0) prev = hi ? p7 : 0.0f;         // i==8 needs P[7][n]; i==0 none
        else        prev = d[r - 1];               // P[i-1][n], same lane
        res[r] = pw * c + prev;
        pw *= oma;
    }

    // Each lane owns 8 consecutive floats: two 16B stores.
    float4* o = reinterpret_cast<float4*>(out + base + (long long)m * 16 + (hi ? 8 : 0));
    o[0] = make_float4(res[0], res[1], res[2], res[3]);
    o[1] = make_float4(res[4], res[5], res[6], res[7]);
}

// ---------------------------------------------------------------------------
extern "C" void kernel_launch(void* const* d_in, const int* in_sizes, int n_in,
                              void* d_out, int out_size, void* d_ws, size_t ws_size,
                              hipStream_t stream) {
    const float* x     = (const float*)d_in[0];
    const float* alpha = (const float*)d_in[1];
    const float* level = (const float*)d_in[2];
    float* out = (float*)d_out;

    const int n      = in_sizes[0];                // 16777216
    const int ntiles = n / TILE;                   // 65536

    float* btile = (float*)d_ws;                   // [ntiles]
    float* carry = btile + ntiles;                 // [ntiles]

    const int blocks = (ntiles + WAVES_PER_BLOCK - 1) / WAVES_PER_BLOCK;

    ses_phase1<<<blocks, BLOCK_THREADS, 0, stream>>>(x, alpha, btile, ntiles);
    ses_phase2<<<1, 32, 0, stream>>>(btile, level, alpha, carry, ntiles);
    ses_phase3<<<blocks, BLOCK_THREADS, 0, stream>>>(x, alpha, carry, out, ntiles);
}